// IndexValue_8134668059088
// MI455X (gfx1250) — compile-verified
//
#include <hip/hip_runtime.h>
#include <stdint.h>

// out[s,a] = values[index[s,a]]  -- pure bandwidth-bound gather.
// CDNA5 path: async global->LDS staging of the 100-entry table
// (global_load_async_to_lds_b128 + s_wait_asynccnt), ds_load gathers,
// 128-bit non-temporal streamed index loads / output stores with
// 32-bit saddr+voffset addressing (all offsets < 2^27 bytes).

typedef int   i32x4 __attribute__((ext_vector_type(4)));
typedef float f32x4 __attribute__((ext_vector_type(4)));

#define THREADS 256
#define ELEMS_PER_THREAD 8   // two b128 loads / two b128 stores per thread

__global__ __launch_bounds__(THREADS) void gather_vocab_kernel(
    const int* __restrict__ index,
    const float* __restrict__ values,
    float* __restrict__ out,
    unsigned n)  // total elements (33,554,432)
{
    // Table padded to 128 entries; only [0,100) valid/used.
    __shared__ float table[128];

    // --- Stage values[0..99] into LDS via CDNA5 async global->LDS copies ---
    // 400 bytes = 25 x 16B transfers, lanes 0..24 of wave 0.
    if (threadIdx.x < 25) {
        // Low 32 bits of a generic LDS pointer = wave-relative LDS byte
        // offset, which is what the async op's VDST VGPR expects.
        unsigned lds_off = (unsigned)(uintptr_t)(&table[threadIdx.x * 4]);
        unsigned long long gaddr =
            (unsigned long long)(uintptr_t)(values + threadIdx.x * 4);
        asm volatile("global_load_async_to_lds_b128 %0, %1, off"
                     :
                     : "v"(lds_off), "v"(gaddr)
                     : "memory");
    }
    // Drain ASYNCcnt before publishing LDS contents to the workgroup.
    asm volatile("s_wait_asynccnt 0x0" ::: "memory");
    __syncthreads();

    const unsigned gid  = blockIdx.x * THREADS + threadIdx.x;
    const unsigned base = gid * (unsigned)ELEMS_PER_THREAD;  // element index

    if (base + ELEMS_PER_THREAD <= n) {
        // Byte offsets fit in u32 (max 2^27) -> saddr + 32-bit voffset VMEM.
        const unsigned byte0 = base * 4u;

        // Streamed 128-bit non-temporal index loads (read-once data).
        i32x4 ia = __builtin_nontemporal_load(
            (const i32x4*)((const char*)index + byte0));
        i32x4 ib = __builtin_nontemporal_load(
            (const i32x4*)((const char*)index + byte0 + 16u));

        // LDS gathers (ds_load_b32 x8) -- DS pipe, VMEM pipe stays free
        // for the streaming loads/stores.
        f32x4 ra, rb;
        ra.x = table[ia.x];
        ra.y = table[ia.y];
        ra.z = table[ia.z];
        ra.w = table[ia.w];
        rb.x = table[ib.x];
        rb.y = table[ib.y];
        rb.z = table[ib.z];
        rb.w = table[ib.w];

        // Streamed 128-bit non-temporal stores (write-once data).
        __builtin_nontemporal_store(ra, (f32x4*)((char*)out + byte0));
        __builtin_nontemporal_store(rb, (f32x4*)((char*)out + byte0 + 16u));
    } else {
        // Tail (never hit for the reference shapes; kept for generality).
        for (unsigned i = base; i < n; ++i) {
            out[i] = table[index[i]];
        }
    }
}

extern "C" void kernel_launch(void* const* d_in, const int* in_sizes, int n_in,
                              void* d_out, int out_size, void* d_ws, size_t ws_size,
                              hipStream_t stream) {
    const int*   index  = (const int*)d_in[0];    // [16384, 2048] int32
    const float* values = (const float*)d_in[1];  // [100] f32
    float*       out    = (float*)d_out;          // [16384, 2048] f32

    const unsigned n = (unsigned)in_sizes[0];     // 33,554,432
    const unsigned n_threads =
        (n + ELEMS_PER_THREAD - 1) / ELEMS_PER_THREAD;
    const unsigned blocks = (n_threads + THREADS - 1) / THREADS;  // 16384

    gather_vocab_kernel<<<blocks, THREADS, 0, stream>>>(index, values, out, n);
}